// AttnDecoder_2010044695221
// MI455X (gfx1250) — compile-verified
//
#include <hip/hip_runtime.h>
#include <hip/hip_bf16.h>
#include <math.h>

// ---------------------------------------------------------------------------
// AttnDecoder single-step, MI455X (gfx1250, wave32).
// Outputs depend only on: emb[tok], h0, c0, W_ih, W_hh, b_ih, b_hh, out_W,
// out_b.  The attention branch in the reference is dead code w.r.t. the
// returned tuple and is skipped (outputs are bit-for-bit unaffected).
// All matvecs run on V_WMMA_F32_16X16X4_F32 (exact f32 math on the matrix
// pipe); the workload is HBM-bound (~237 MB weights => ~10 us @ 23.3 TB/s).
// ---------------------------------------------------------------------------

typedef __attribute__((ext_vector_type(2))) float v2f;
typedef __attribute__((ext_vector_type(8))) float v8f;

#define HD   1024           // H
#define VSZ  50000          // V
#define G4H  4096           // 4*H

// Accumulate acc += W_tile(16 rows x HD) @ vec, where `vec` is an LDS copy of
// the activation vector.  Mapping: A = vec broadcast to all 16 M-rows
// (per-lane frag: lanes 0-15 -> K {k0,k0+1}, lanes 16-31 -> K {k0+2,k0+3});
// B = weight tile, B[k][n] = W[row0+n][k] (lane n&15 supplies its row's
// contiguous float2 at the lane-half's K offset).  After the loop every lane
// holds the dot product for column (lane&15) in acc[0].
__device__ __forceinline__ void wmma_matvec_1024(const float* __restrict__ wrow,
                                                 const float* __restrict__ svec,
                                                 int koff, v8f& acc)
{
#pragma unroll 8
    for (int k0 = 0; k0 < HD; k0 += 4) {
        v2f a = *(const v2f*)(svec + k0 + koff);   // ds_load_b64 (shared vec)
        v2f b = *(const v2f*)(wrow + k0 + koff);   // global_load_b64 (weights)
        acc = __builtin_amdgcn_wmma_f32_16x16x4_f32(
            /*neg_a=*/false, a, /*neg_b=*/false, b,
            /*c_mod=*/(short)0, acc, /*reuse_a=*/false, /*reuse_b=*/false);
    }
}

// ---- Stage 1: gates[4096] = W_ih @ e + W_hh @ h0 ---------------------------
__global__ void __launch_bounds__(256)
lstm_gates_wmma(const float* __restrict__ W_ih, const float* __restrict__ W_hh,
                const float* __restrict__ emb,  const int*   __restrict__ tok,
                const float* __restrict__ h0,   float* __restrict__ gates)
{
    __shared__ __align__(16) float svec[2 * HD];   // [e | h]
    const float* e = emb + (size_t)tok[0] * HD;
    for (int i = threadIdx.x; i < HD; i += blockDim.x) {
        svec[i]      = e[i];
        svec[HD + i] = h0[i];
    }
    __syncthreads();

    const int lane  = threadIdx.x & 31;
    const int half  = lane >> 4;                   // 0: K {0,1}, 1: K {2,3}
    const int koff  = 2 * half;
    const int wid   = blockIdx.x * (blockDim.x >> 5) + (threadIdx.x >> 5);
    const int nwv   = gridDim.x * (blockDim.x >> 5);

    for (int tile = wid; tile < G4H / 16; tile += nwv) {
        const int row0 = tile * 16;
        const size_t roff = (size_t)(row0 + (lane & 15)) * HD;
        v8f acc = {0.f, 0.f, 0.f, 0.f, 0.f, 0.f, 0.f, 0.f};
        wmma_matvec_1024(W_ih + roff, svec,      koff, acc);
        wmma_matvec_1024(W_hh + roff, svec + HD, koff, acc);
        if (lane < 16) gates[row0 + lane] = acc[0];
    }
}

// ---- Stage 2: LSTM pointwise -> h_new, c_new into d_out[V:] ----------------
__global__ void __launch_bounds__(256)
lstm_pointwise(const float* __restrict__ gates, const float* __restrict__ b_ih,
               const float* __restrict__ b_hh,  const float* __restrict__ c0,
               float* __restrict__ out)
{
    const int t = blockIdx.x * blockDim.x + threadIdx.x;
    if (t >= HD) return;
    const float gi = gates[t]          + b_ih[t]          + b_hh[t];
    const float gf = gates[HD + t]     + b_ih[HD + t]     + b_hh[HD + t];
    const float gg = gates[2 * HD + t] + b_ih[2 * HD + t] + b_hh[2 * HD + t];
    const float go = gates[3 * HD + t] + b_ih[3 * HD + t] + b_hh[3 * HD + t];
    const float si = 1.f / (1.f + __expf(-gi));
    const float sf = 1.f / (1.f + __expf(-gf));
    const float so = 1.f / (1.f + __expf(-go));
    const float cn = sf * c0[t] + si * tanhf(gg);
    const float hn = so * tanhf(cn);
    out[VSZ + t]      = hn;   // h_new
    out[VSZ + HD + t] = cn;   // c_new
}

// ---- Stage 3: logits[50000] = out_W @ h_new + out_b ------------------------
__global__ void __launch_bounds__(256)
logits_wmma(const float* __restrict__ outW, const float* __restrict__ outB,
            const float* __restrict__ hvec, float* __restrict__ logits)
{
    __shared__ __align__(16) float svec[HD];
    for (int i = threadIdx.x; i < HD; i += blockDim.x) svec[i] = hvec[i];
    __syncthreads();

    const int lane = threadIdx.x & 31;
    const int koff = 2 * (lane >> 4);
    const int wid  = blockIdx.x * (blockDim.x >> 5) + (threadIdx.x >> 5);
    const int nwv  = gridDim.x * (blockDim.x >> 5);

    for (int tile = wid; tile < VSZ / 16; tile += nwv) {
        const int row0 = tile * 16;
        v8f acc = {0.f, 0.f, 0.f, 0.f, 0.f, 0.f, 0.f, 0.f};
        wmma_matvec_1024(outW + (size_t)(row0 + (lane & 15)) * HD, svec, koff, acc);
        if (lane < 16) logits[row0 + lane] = acc[0] + outB[row0 + lane];
    }
}

// ---- Stage 4: log-softmax statistics (max, log-sum-exp) --------------------
__global__ void __launch_bounds__(1024)
logsoftmax_reduce(const float* __restrict__ logits, float* __restrict__ stats)
{
    __shared__ float red[1024];
    float m = -INFINITY;
    for (int i = threadIdx.x; i < VSZ; i += blockDim.x) m = fmaxf(m, logits[i]);
    red[threadIdx.x] = m;
    __syncthreads();
    for (int s = 512; s > 0; s >>= 1) {
        if (threadIdx.x < s)
            red[threadIdx.x] = fmaxf(red[threadIdx.x], red[threadIdx.x + s]);
        __syncthreads();
    }
    m = red[0];
    __syncthreads();

    float sum = 0.f;
    for (int i = threadIdx.x; i < VSZ; i += blockDim.x) sum += __expf(logits[i] - m);
    red[threadIdx.x] = sum;
    __syncthreads();
    for (int s = 512; s > 0; s >>= 1) {
        if (threadIdx.x < s) red[threadIdx.x] += red[threadIdx.x + s];
        __syncthreads();
    }
    if (threadIdx.x == 0) { stats[0] = m; stats[1] = __logf(red[0]); }
}

// ---- Stage 5: apply --------------------------------------------------------
__global__ void __launch_bounds__(256)
logsoftmax_apply(const float* __restrict__ logits, const float* __restrict__ stats,
                 float* __restrict__ out)
{
    const int i = blockIdx.x * blockDim.x + threadIdx.x;
    if (i < VSZ) out[i] = logits[i] - stats[0] - stats[1];
}

// ---------------------------------------------------------------------------
extern "C" void kernel_launch(void* const* d_in, const int* in_sizes, int n_in,
                              void* d_out, int out_size, void* d_ws, size_t ws_size,
                              hipStream_t stream)
{
    (void)in_sizes; (void)n_in; (void)out_size; (void)ws_size;

    const int*   tok   = (const int*)  d_in[0];   // int64 low word (LE) is fine
    const float* h0    = (const float*)d_in[1];
    const float* c0    = (const float*)d_in[2];
    const float* emb   = (const float*)d_in[4];
    const float* W_ih  = (const float*)d_in[9];
    const float* W_hh  = (const float*)d_in[10];
    const float* b_ih  = (const float*)d_in[11];
    const float* b_hh  = (const float*)d_in[12];
    const float* out_W = (const float*)d_in[13];
    const float* out_b = (const float*)d_in[14];

    float* out    = (float*)d_out;
    float* ws     = (float*)d_ws;
    float* gates  = ws;               // 4096 floats
    float* logits = ws + G4H;         // 50000 floats
    float* stats  = ws + G4H + VSZ;   // 2 floats

    // 4096 rows -> 256 tiles of 16 rows; 8 waves/block -> 32 blocks
    lstm_gates_wmma<<<32, 256, 0, stream>>>(W_ih, W_hh, emb, tok, h0, gates);
    lstm_pointwise<<<(HD + 255) / 256, 256, 0, stream>>>(gates, b_ih, b_hh, c0, out);
    // 50000 rows -> 3125 tiles; 392 blocks * 8 waves = 3136 waves
    logits_wmma<<<392, 256, 0, stream>>>(out_W, out_b, out + VSZ, logits);
    logsoftmax_reduce<<<1, 1024, 0, stream>>>(logits, stats);
    logsoftmax_apply<<<(VSZ + 255) / 256, 256, 0, stream>>>(logits, stats, out);
}